// SimpleNet_22986664968457
// MI455X (gfx1250) — compile-verified
//
#include <hip/hip_runtime.h>

#define NN  50000
#define FF  128
#define EE  800000
#define EEI 400000
#define GG  64
#define WUC 129   // wu is [128][129] row-major

typedef __attribute__((ext_vector_type(2))) float v2f;
typedef __attribute__((ext_vector_type(8))) float v8f;

// ---------------------------------------------------------------- zero
__global__ void zero_kernel(float* __restrict__ p, int n) {
  int i = blockIdx.x * blockDim.x + threadIdx.x;
  if (i < n) p[i] = 0.0f;
}

// ----------------------------------------------- degree / count histograms
__global__ void hist_kernel(const int* __restrict__ esrc,
                            const int* __restrict__ isrc,
                            const int* __restrict__ batch,
                            float* __restrict__ degE,
                            float* __restrict__ degI,
                            float* __restrict__ cnt) {
  int i = blockIdx.x * blockDim.x + threadIdx.x;
  if (i < EE) {
    atomicAdd(&degE[esrc[i]], 1.0f);
  } else if (i < EE + EEI) {
    atomicAdd(&degI[isrc[i - EE]], 1.0f);
  } else if (i < EE + EEI + NN) {
    atomicAdd(&cnt[batch[i - EE - EEI]], 1.0f);
  }
}

// ------------------------------------------------------------- fused conv
// out = row_softmax( leaky_relu( xin @ wu[:, :128]^T + deg * wu[:,128] + bu ) )
// optionally writes xout [N,128]; optionally atomic-adds into pool[batch[n]].
__global__ __launch_bounds__(256)
void conv_kernel(const float* __restrict__ xin,
                 const float* __restrict__ deg,
                 const float* __restrict__ wu,    // [128][129]
                 const float* __restrict__ bu,    // [128]
                 float* __restrict__ xout,        // may be null
                 float* __restrict__ pool,        // may be null; [GG][256]
                 const int* __restrict__ batch,
                 int pool_off) {
  __shared__ float s_tile[16 * 132];
  __shared__ float s_wcol[128];
  __shared__ float s_bias[128];
  __shared__ float s_red[16][16];

  const int tid  = threadIdx.x;
  const int lane = tid & 31;     // wave32
  const int wid  = tid >> 5;     // 8 waves, wave w owns cols 16w..16w+15

  if (tid < 128) {
    s_wcol[tid] = wu[tid * WUC + 128];
    s_bias[tid] = bu[tid];
  }

  // Preload B fragments (W^T, all K for this wave's 16 columns) into VGPRs.
  // B layout mirrors C: v0 = K=k0 (lanes0-15) / K=k0+2 (lanes16-31), v1 = k0+1 / k0+3.
  const int ncol = wid * 16 + (lane & 15);
  const int koff = (lane >> 4) << 1;  // 0 or 2
  v2f bfrag[32];
  const float* wp = wu + ncol * WUC + koff;
#pragma unroll
  for (int s = 0; s < 32; ++s) {
    v2f b; b.x = wp[4 * s]; b.y = wp[4 * s + 1];
    bfrag[s] = b;
  }
  __syncthreads();

  const int ntiles = NN / 16;  // 3125, exact
  for (int tile = blockIdx.x; tile < ntiles; tile += gridDim.x) {
    const int row0 = tile * 16;

    // A fragment (16x4 f32): lanes0-15 row M=lane K=k0..k0+1; lanes16-31 K=k0+2..k0+3
    v8f c = {};
    const float* xp = xin + (size_t)(row0 + (lane & 15)) * FF + koff;
#pragma unroll
    for (int s = 0; s < 32; ++s) {
      v2f a; a.x = xp[4 * s]; a.y = xp[4 * s + 1];
      c = __builtin_amdgcn_wmma_f32_16x16x4_f32(false, a, false, bfrag[s],
                                                (short)0, c, false, false);
    }

    // Spill accumulator tile to LDS (VGPR r: row r for lanes0-15, row r+8 else)
    {
      const int col = wid * 16 + (lane & 15);
      const int rb  = (lane >> 4) * 8;
#pragma unroll
      for (int r = 0; r < 8; ++r)
        s_tile[(rb + r) * 132 + col] = c[r];
    }
    __syncthreads();

    // Fused epilogue: +deg*w_col + bias -> leaky_relu -> row softmax.
    // 16 threads per row, 8 cols each.
    const int row = tid >> 4;
    const int sub = tid & 15;
    const float dg = deg[row0 + row];
    float vals[8];
    float m = -3.402823466e38f;
#pragma unroll
    for (int j = 0; j < 8; ++j) {
      const int col = sub * 8 + j;
      float v = s_tile[row * 132 + col] + dg * s_wcol[col] + s_bias[col];
      v = (v > 0.0f) ? v : 0.01f * v;
      vals[j] = v;
      m = fmaxf(m, v);
    }
    s_red[row][sub] = m;
    __syncthreads();
    float rmax = -3.402823466e38f;
#pragma unroll
    for (int i = 0; i < 16; ++i) rmax = fmaxf(rmax, s_red[row][i]);
    float esum = 0.0f;
#pragma unroll
    for (int j = 0; j < 8; ++j) { vals[j] = __expf(vals[j] - rmax); esum += vals[j]; }
    __syncthreads();
    s_red[row][sub] = esum;
    __syncthreads();
    float rsum = 0.0f;
#pragma unroll
    for (int i = 0; i < 16; ++i) rsum += s_red[row][i];
    const float inv = 1.0f / rsum;
#pragma unroll
    for (int j = 0; j < 8; ++j) vals[j] *= inv;

    if (xout) {
      float* op = xout + (size_t)(row0 + row) * FF + sub * 8;
#pragma unroll
      for (int j = 0; j < 8; ++j) op[j] = vals[j];
    }
    if (pool) {
      const int g = batch[row0 + row];
      float* pp = pool + g * 256 + pool_off + sub * 8;
#pragma unroll
      for (int j = 0; j < 8; ++j) atomicAdd(pp + j, vals[j]);
    }
    __syncthreads();  // protect s_tile/s_red reuse next tile
  }
}

// -------------------------------------------------- pooled-feature MLP head
__global__ __launch_bounds__(128)
void mlp_kernel(const float* __restrict__ pool,   // [GG][256] sums
                const float* __restrict__ cnt,    // [GG]
                const float* __restrict__ fc1_w,  // [128][256]
                const float* __restrict__ fc1_b,  // [128]
                const float* __restrict__ fc2_w,  // [1][128]
                const float* __restrict__ fc2_b,  // [1]
                float* __restrict__ out) {        // [GG]
  __shared__ float zin[256];
  __shared__ float zred[128];
  const int g = blockIdx.x;
  const int t = threadIdx.x;
  const float c = fmaxf(cnt[g], 1.0f);
  for (int i = t; i < 256; i += 128) zin[i] = pool[g * 256 + i] / c;
  __syncthreads();
  float acc = fc1_b[t];
  const float* wr = fc1_w + t * 256;
#pragma unroll 8
  for (int i = 0; i < 256; ++i) acc += zin[i] * wr[i];
  acc = fmaxf(acc, 0.0f);
  zred[t] = acc * fc2_w[t];
  __syncthreads();
  for (int s = 64; s > 0; s >>= 1) {
    if (t < s) zred[t] += zred[t + s];
    __syncthreads();
  }
  if (t == 0) out[g] = zred[0] + fc2_b[0];
}

// ---------------------------------------------------------------- launcher
extern "C" void kernel_launch(void* const* d_in, const int* in_sizes, int n_in,
                              void* d_out, int out_size, void* d_ws, size_t ws_size,
                              hipStream_t stream) {
  (void)in_sizes; (void)n_in; (void)out_size; (void)ws_size;
  const float* x     = (const float*)d_in[0];
  const int*   eidx  = (const int*)d_in[1];   // [2,E]; row 0 = sources
  const int*   ieidx = (const int*)d_in[3];   // [2,EI]
  const int*   batch = (const int*)d_in[5];
  const float* wu_e1 = (const float*)d_in[8],  *bu_e1 = (const float*)d_in[9];
  const float* wu_e2 = (const float*)d_in[12], *bu_e2 = (const float*)d_in[13];
  const float* wu_i1 = (const float*)d_in[16], *bu_i1 = (const float*)d_in[17];
  const float* wu_i2 = (const float*)d_in[20], *bu_i2 = (const float*)d_in[21];
  const float* fc1_w = (const float*)d_in[22], *fc1_b = (const float*)d_in[23];
  const float* fc2_w = (const float*)d_in[24], *fc2_b = (const float*)d_in[25];

  float* ws   = (float*)d_ws;
  float* degE = ws;
  float* degI = degE + NN;
  float* cnt  = degI + NN;
  float* pool = cnt + GG;
  float* ext1 = pool + GG * 256;
  float* itn1 = ext1 + (size_t)NN * FF;

  const int zn = 2 * NN + GG + GG * 256;
  zero_kernel<<<(zn + 255) / 256, 256, 0, stream>>>(ws, zn);

  const int hn = EE + EEI + NN;
  hist_kernel<<<(hn + 255) / 256, 256, 0, stream>>>(eidx, ieidx, batch, degE, degI, cnt);

  const int cgrid = NN / 16;  // 3125 row tiles
  conv_kernel<<<cgrid, 256, 0, stream>>>(x,    degE, wu_e1, bu_e1, ext1,    nullptr, batch, 0);
  conv_kernel<<<cgrid, 256, 0, stream>>>(ext1, degE, wu_e2, bu_e2, nullptr, pool,    batch, 0);
  conv_kernel<<<cgrid, 256, 0, stream>>>(x,    degI, wu_i1, bu_i1, itn1,    nullptr, batch, 0);
  conv_kernel<<<cgrid, 256, 0, stream>>>(itn1, degI, wu_i2, bu_i2, nullptr, pool,    batch, 128);

  mlp_kernel<<<GG, 128, 0, stream>>>(pool, cnt, fc1_w, fc1_b, fc2_w, fc2_b, (float*)d_out);
}